// STGCNModel_14216341750333
// MI455X (gfx1250) — compile-verified
//
#include <hip/hip_runtime.h>
#include <hip/hip_bf16.h>
#include <math.h>

#define Bn   128
#define Fn   512
#define Nn   256
#define Dn   2
#define En   512           // 2*N
#define COUT 8
#define HIDn 64
#define Cn   100
#define EPSc 1e-5f
#define PAIRS (Bn*Fn*Nn)   // 16,777,216 float2 pairs
#define NPART 1024

typedef float v2f __attribute__((ext_vector_type(2)));
typedef float v8f __attribute__((ext_vector_type(8)));

// ---------------- BatchNorm statistics (2-stage reduction) ----------------
__global__ void bn_stats_partial(const float* __restrict__ x, float* __restrict__ part) {
  float s0 = 0.f, s1 = 0.f, q0 = 0.f, q1 = 0.f;
  const float2* xp = (const float2*)x;
  for (int i = blockIdx.x * blockDim.x + threadIdx.x; i < PAIRS; i += gridDim.x * blockDim.x) {
    float2 v = xp[i];
    s0 += v.x; s1 += v.y; q0 += v.x * v.x; q1 += v.y * v.y;
  }
  __shared__ float sm[256][4];
  int t = threadIdx.x;
  sm[t][0] = s0; sm[t][1] = s1; sm[t][2] = q0; sm[t][3] = q1;
  __syncthreads();
  for (int s = 128; s > 0; s >>= 1) {
    if (t < s) { sm[t][0] += sm[t+s][0]; sm[t][1] += sm[t+s][1];
                 sm[t][2] += sm[t+s][2]; sm[t][3] += sm[t+s][3]; }
    __syncthreads();
  }
  if (t == 0) {
    part[blockIdx.x*4+0] = sm[0][0]; part[blockIdx.x*4+1] = sm[0][1];
    part[blockIdx.x*4+2] = sm[0][2]; part[blockIdx.x*4+3] = sm[0][3];
  }
}

__global__ void bn_stats_final(const float* __restrict__ part, const float* __restrict__ gamma,
                               const float* __restrict__ beta, float* __restrict__ stats) {
  __shared__ float sm[256][4];
  int t = threadIdx.x;
  float s0 = 0.f, s1 = 0.f, q0 = 0.f, q1 = 0.f;
  for (int i = t; i < NPART; i += 256) {
    s0 += part[i*4+0]; s1 += part[i*4+1]; q0 += part[i*4+2]; q1 += part[i*4+3];
  }
  sm[t][0] = s0; sm[t][1] = s1; sm[t][2] = q0; sm[t][3] = q1;
  __syncthreads();
  for (int s = 128; s > 0; s >>= 1) {
    if (t < s) { sm[t][0] += sm[t+s][0]; sm[t][1] += sm[t+s][1];
                 sm[t][2] += sm[t+s][2]; sm[t][3] += sm[t+s][3]; }
    __syncthreads();
  }
  if (t == 0) {
    const float cnt = (float)PAIRS;
    float m0 = sm[0][0] / cnt, m1 = sm[0][1] / cnt;
    float v0 = sm[0][2] / cnt - m0 * m0;
    float v1 = sm[0][3] / cnt - m1 * m1;
    float sc0 = gamma[0] * rsqrtf(v0 + EPSc);
    float sc1 = gamma[1] * rsqrtf(v1 + EPSc);
    stats[0] = sc0; stats[1] = sc1;
    stats[2] = beta[0] - m0 * sc0;
    stats[3] = beta[1] - m1 * sc1;
  }
}

// ---------------- q = var@Wq+bq ; qk = Wk@q ; fold BN affine + 1/sqrt(E) ----------------
__global__ void qk_kernel(const float* __restrict__ variance, const float* __restrict__ Wq,
                          const float* __restrict__ bq, const float* __restrict__ Wk,
                          const float* __restrict__ bk, const float* __restrict__ stats,
                          float* __restrict__ qkp, float* __restrict__ cbuf) {
  __shared__ float vrow[En];
  __shared__ float qsh[En];
  __shared__ float red[En];
  int b = blockIdx.x, t = threadIdx.x;   // 512 threads
  vrow[t] = variance[b * En + t];
  __syncthreads();
  float acc = bq[t];
  for (int e = 0; e < En; ++e) acc += vrow[e] * Wq[e * En + t];
  qsh[t] = acc;
  __syncthreads();
  float qk = 0.f;
  const float* wkrow = Wk + (size_t)t * En;
  for (int e = 0; e < En; ++e) qk += wkrow[e] * qsh[e];
  const float inv = 0.04419417382415922f;  // 1/sqrt(512)
  float scale = stats[t & 1];
  float shift = stats[2 + (t & 1)];
  qkp[b * En + t] = scale * qk * inv;
  red[t] = shift * qk + qsh[t] * bk[t];
  __syncthreads();
  for (int s = 256; s > 0; s >>= 1) { if (t < s) red[t] += red[t + s]; __syncthreads(); }
  if (t == 0) cbuf[b] = red[0] * inv;
}

// ---------------- scores[b,f] = features[b,f,:] . qk'[b] + c[b] (one wave per row) --------
__global__ void scores_kernel(const float* __restrict__ x, const float* __restrict__ qkp,
                              const float* __restrict__ cbuf, float* __restrict__ scores) {
  int wid  = (blockIdx.x * blockDim.x + threadIdx.x) >> 5;  // row in [0, B*F)
  int lane = threadIdx.x & 31;
  int b = wid >> 9;                                          // /F
  const float4* xr = (const float4*)(x   + (size_t)wid * En);
  const float4* qr = (const float4*)(qkp + (size_t)b  * En);
  float acc = 0.f;
#pragma unroll
  for (int k = 0; k < 4; ++k) {
    float4 xv = xr[lane + 32 * k];
    float4 qv = qr[lane + 32 * k];
    acc += xv.x * qv.x + xv.y * qv.y + xv.z * qv.z + xv.w * qv.w;
  }
#pragma unroll
  for (int off = 16; off > 0; off >>= 1) acc += __shfl_xor(acc, off);
  if (lane == 0) scores[wid] = acc + cbuf[b];
}

// ---------------- softmax over F per batch ----------------
__global__ void softmax_f(const float* __restrict__ scores, float* __restrict__ attn) {
  __shared__ float sm[256];
  int b = blockIdx.x, t = threadIdx.x;
  float s0 = scores[b * Fn + t];
  float s1 = scores[b * Fn + t + 256];
  sm[t] = fmaxf(s0, s1);
  __syncthreads();
  for (int s = 128; s > 0; s >>= 1) { if (t < s) sm[t] = fmaxf(sm[t], sm[t + s]); __syncthreads(); }
  float mx = sm[0];
  __syncthreads();
  float e0 = expf(s0 - mx), e1 = expf(s1 - mx);
  sm[t] = e0 + e1;
  __syncthreads();
  for (int s = 128; s > 0; s >>= 1) { if (t < s) sm[t] += sm[t + s]; __syncthreads(); }
  float invs = 1.f / sm[0];
  attn[b * Fn + t]       = e0 * invs;
  attn[b * Fn + t + 256] = e1 * invs;
}

// ---------------- z[b,e] = scale*(sum_f attn*X) + shift ----------------
// Attention row staged to LDS via gfx1250 async global->LDS copy (ASYNCcnt path);
// feature rows prefetched ahead with global_prefetch_b8.
__global__ void wsum_kernel(const float* __restrict__ x, const float* __restrict__ attn,
                            const float* __restrict__ stats, float* __restrict__ z) {
  __shared__ float a[Fn];
  int b = blockIdx.x, t = threadIdx.x;   // 256 threads
  {
    // Each lane copies one float2 of the attention row straight into LDS.
    unsigned lds = (unsigned)(unsigned long long)&a[2 * t];        // low 32 bits = LDS byte addr
    unsigned long long ga = (unsigned long long)(attn + (size_t)b * Fn + 2 * t);
    asm volatile("global_load_async_to_lds_b64 %0, %1, off"
                 :: "v"(lds), "v"(ga) : "memory");
    asm volatile("s_wait_asynccnt 0" ::: "memory");
  }
  __syncthreads();
  const float* xb = x + (size_t)b * Fn * En;
  float acc0 = 0.f, acc1 = 0.f;
#pragma unroll 8
  for (int f = 0; f < Fn; ++f) {
    if ((f & 7) == 0) {
      // Prefetch a row ~64 iterations ahead; wrap mask keeps the address valid & branchless.
      __builtin_prefetch(xb + (size_t)((f + 64) & (Fn - 1)) * En + t, 0, 0);
    }
    float af = a[f];
    acc0 += af * xb[(size_t)f * En + t];
    acc1 += af * xb[(size_t)f * En + t + 256];
  }
  z[b * En + t]       = stats[t & 1] * acc0 + stats[2 + (t & 1)];
  z[b * En + t + 256] = stats[t & 1] * acc1 + stats[2 + (t & 1)];
}

// ---------------- fp32 WMMA GEMM: C(M,N) = A(M,K) @ W(K,N) + bias(N) ----------------
// One 16x16 output tile per wave via V_WMMA_F32_16X16X4_F32; 8 waves/block = 32x64 tile.
// Column handling is branchless: out-of-range lanes clamp their load address and compute
// garbage that is never stored (WMMA column n only consumes B-fragment lanes of column n).
__global__ void gemm_wmma(const float* __restrict__ A, const float* __restrict__ W,
                          const float* __restrict__ bias, float* __restrict__ Cmat,
                          int M, int N, int K) {
  int wave = threadIdx.x >> 5;
  int lane = threadIdx.x & 31;
  int half = lane >> 4;      // 0: K pair {0,1}, 1: K pair {2,3}
  int l16  = lane & 15;
  int m0 = blockIdx.y * 32 + (wave >> 2) * 16;
  int n0 = blockIdx.x * 64 + (wave & 3) * 16;
  int row = m0 + l16;
  int col = n0 + l16;
  int colc = (col < N) ? col : (N - 1);                  // loop-invariant clamp
  const float* ap = A + (size_t)row * K + half * 2;
  const float* wp = W + (size_t)(half * 2) * N + colc;
  v8f c = {};
#pragma unroll 8
  for (int k0 = 0; k0 < K; k0 += 4) {
    v2f a, bm;
    a[0]  = ap[0];
    a[1]  = ap[1];
    bm[0] = wp[0];
    bm[1] = wp[N];
    c = __builtin_amdgcn_wmma_f32_16x16x4_f32(false, a, false, bm, (short)0, c, false, false);
    ap += 4;
    wp += (size_t)4 * N;
  }
  if (col < N) {
    float bc = bias[col];
#pragma unroll
    for (int r = 0; r < 8; ++r)
      Cmat[(size_t)(m0 + r + half * 8) * N + col] = c[r] + bc;
  }
}

// ---------------- spatial: g[b,n,:] = ((A+I)/deg @ y)[n] @ Wconv (one wave per row) -------
__global__ void spatial_kernel(const float* __restrict__ adj, const float* __restrict__ y,
                               const float* __restrict__ Wconv, float* __restrict__ g) {
  int wid  = (blockIdx.x * blockDim.x + threadIdx.x) >> 5;  // [0, B*N)
  int lane = threadIdx.x & 31;
  int b = wid >> 8;           // /N
  int n = wid & 255;
  const float* Ar = adj + (size_t)wid * Nn;
  const float* Y  = y   + (size_t)b * En;
  float deg = 0.f, s0 = 0.f, s1 = 0.f;
  for (int m = lane; m < Nn; m += 32) {
    float a = Ar[m] + ((m == n) ? 1.f : 0.f);   // self loop
    deg += a;
    s0 += a * Y[2 * m];
    s1 += a * Y[2 * m + 1];
  }
#pragma unroll
  for (int off = 16; off > 0; off >>= 1) {
    deg += __shfl_xor(deg, off);
    s0  += __shfl_xor(s0, off);
    s1  += __shfl_xor(s1, off);
  }
  if (lane == 0) {
    float inv = 1.f / deg;
    float g0 = s0 * inv, g1 = s1 * inv;
#pragma unroll
    for (int j = 0; j < COUT; ++j)
      g[(size_t)wid * COUT + j] = g0 * Wconv[j] + g1 * Wconv[COUT + j];
  }
}

// ---------------- final row softmax over C=100 ----------------
__global__ void softmax_out(const float* __restrict__ lg, float* __restrict__ out) {
  __shared__ float sm[128];
  int b = blockIdx.x, t = threadIdx.x;
  float v = (t < Cn) ? lg[b * Cn + t] : -3.0e38f;
  sm[t] = v;
  __syncthreads();
  for (int s = 64; s > 0; s >>= 1) { if (t < s) sm[t] = fmaxf(sm[t], sm[t + s]); __syncthreads(); }
  float mx = sm[0];
  __syncthreads();
  float e = (t < Cn) ? expf(v - mx) : 0.f;
  sm[t] = e;
  __syncthreads();
  for (int s = 64; s > 0; s >>= 1) { if (t < s) sm[t] += sm[t + s]; __syncthreads(); }
  if (t < Cn) out[b * Cn + t] = e / sm[0];
}

static inline void launch_gemm(const float* A, const float* W, const float* bias, float* Cm,
                               int M, int N, int K, hipStream_t s) {
  dim3 grid((N + 63) / 64, (M + 31) / 32);
  gemm_wmma<<<grid, 256, 0, s>>>(A, W, bias, Cm, M, N, K);
}

extern "C" void kernel_launch(void* const* d_in, const int* in_sizes, int n_in,
                              void* d_out, int out_size, void* d_ws, size_t ws_size,
                              hipStream_t stream) {
  const float* adj   = (const float*)d_in[0];
  const float* feat  = (const float*)d_in[1];
  const float* var   = (const float*)d_in[2];
  const float* gamma = (const float*)d_in[3];
  const float* beta  = (const float*)d_in[4];
  const float* Wq    = (const float*)d_in[5];
  const float* Wk    = (const float*)d_in[6];
  const float* Wv    = (const float*)d_in[7];
  const float* bq    = (const float*)d_in[8];
  const float* bk    = (const float*)d_in[9];
  const float* bv    = (const float*)d_in[10];
  const float* Wo    = (const float*)d_in[11];
  const float* bo    = (const float*)d_in[12];
  const float* Wconv = (const float*)d_in[13];
  const float* W1    = (const float*)d_in[14];
  const float* b1    = (const float*)d_in[15];
  const float* W2    = (const float*)d_in[16];
  const float* b2    = (const float*)d_in[17];
  (void)in_sizes; (void)n_in; (void)out_size; (void)ws_size;

  float* ws = (float*)d_ws;
  size_t o = 0;
  float* part   = ws + o; o += NPART * 4;      // block partials
  float* stats  = ws + o; o += 16;             // scale0,scale1,shift0,shift1
  float* cbuf   = ws + o; o += 128;            // per-batch score constant
  float* qkp    = ws + o; o += Bn * En;        // folded Wk@q vectors
  float* scores = ws + o; o += Bn * Fn;
  float* attn   = ws + o; o += Bn * Fn;
  float* z      = ws + o; o += Bn * En;
  float* t1     = ws + o; o += Bn * En;
  float* t2     = ws + o; o += Bn * En;
  float* g      = ws + o; o += Bn * Nn * COUT; // graph-conv output (B, 2048)
  float* h      = ws + o; o += Bn * HIDn;
  float* lgts   = ws + o; o += Bn * Cn;

  // 1) BatchNorm batch statistics (134 MB streaming reduction)
  bn_stats_partial<<<NPART, 256, 0, stream>>>(feat, part);
  bn_stats_final<<<1, 256, 0, stream>>>(part, gamma, beta, stats);

  // 2) Fold attention K-projection into a per-batch vector qk' + constant c
  qk_kernel<<<Bn, 512, 0, stream>>>(var, Wq, bq, Wk, bk, stats, qkp, cbuf);

  // 3) scores = features . qk' + c  (one wave per (b,f) row; 134 MB pass)
  scores_kernel<<<(Bn * Fn) / 8, 256, 0, stream>>>(feat, qkp, cbuf, scores);

  // 4) softmax over F
  softmax_f<<<Bn, 256, 0, stream>>>(scores, attn);

  // 5) z = BN-affine(attn @ features)  (second 134 MB pass, L2-resident)
  wsum_kernel<<<Bn, 256, 0, stream>>>(feat, attn, stats, z);

  // 6) attention output projections via fp32 WMMA
  launch_gemm(z,  Wv, bv, t1, Bn, En, En, stream);   // (128,512)@(512,512)
  launch_gemm(t1, Wo, bo, t2, Bn, En, En, stream);   // (128,512)@(512,512)

  // 7) spatial graph conv: D^{-1}(A+I) @ y @ Wconv
  spatial_kernel<<<(Bn * Nn) / 8, 256, 0, stream>>>(adj, t2, Wconv, g);

  // 8) classification head via fp32 WMMA
  launch_gemm(g, W1, b1, h, Bn, HIDn, Nn * COUT, stream);  // (128,2048)@(2048,64)
  launch_gemm(h, W2, b2, lgts, Bn, Cn, HIDn, stream);      // (128,64)@(64,100)

  // 9) final softmax
  softmax_out<<<Bn, 128, 0, stream>>>(lgts, (float*)d_out);
}